// JointBilateralFilter_65721589563615
// MI455X (gfx1250) — compile-verified
//
#include <hip/hip_runtime.h>
#include <stdint.h>

// Joint bilateral filter, MI455X (gfx1250).
// VALU/transcendental-bound stencil; WMMA does not map (per-pixel weights ->
// no shared GEMM operand). CDNA5 features used: Tensor Data Mover async tile
// staging into LDS (TENSORcnt + s_wait_tensorcnt) for interior tiles, wave32
// tiled compute.
//
// Guidance is stored in LDS pre-scaled by sqrt(sigma_inv * 0.5*log2(e)) so the
// per-tap Mahalanobis term is just 8x(sub+fma) and the weight is a single
// v_exp_f32 with a free neg source modifier: w = exp2(-sum (u_c - uc_c)^2).

#define RADIUS 5
#define KSIZE  11
#define TILE   16
#define HALO   (TILE + 2 * RADIUS)      // 26
#define LDS_STRIDE 32                   // floats per halo row in LDS (bank-friendly)
#define PLANE_ELEMS (HALO * LDS_STRIDE) // 832 floats per plane
#define NPLANES 11                      // 8 guidance + 3 image
#define H 256
#define W 256
#define GCH 8
#define ICH 3

#if __has_builtin(__builtin_amdgcn_tensor_load_to_lds) && __has_builtin(__builtin_amdgcn_s_wait_tensorcnt)
#define HAVE_TDM 1
#else
#define HAVE_TDM 0
#endif

typedef unsigned int v4u __attribute__((ext_vector_type(4)));
typedef int          v8i __attribute__((ext_vector_type(8)));
typedef int          v4i __attribute__((ext_vector_type(4)));

// fscale[c] = sqrt((1/SIGMA_DIAG[c]) * 0.5 * log2(e))
//   SIGMA_DIAG = {10,10,.02,.02,.02,.1,.1,.1} -> sigma_inv = {.1,.1,50,50,50,10,10,10}
__constant__ const float kFscale[GCH] = {
    0.26857914745f, 0.26857914745f,
    6.00561206f,    6.00561206f,    6.00561206f,
    2.68579147f,    2.68579147f,    2.68579147f
};

__device__ __forceinline__ int reflect_idx(int i) {
    i = (i < 0) ? -i : i;
    return (i > (H - 1)) ? (2 * (H - 1) - i) : i;
}

__global__ __launch_bounds__(256)
void jbf_kernel(const float* __restrict__ image,
                const float* __restrict__ guidance,
                float* __restrict__ out) {
    __shared__ float smem[NPLANES * PLANE_ELEMS]; // 36,608 bytes

    const int tid   = threadIdx.x;
    const int tx    = tid & (TILE - 1);
    const int ty    = tid >> 4;
    const int xbase = blockIdx.x * TILE;
    const int ybase = blockIdx.y * TILE;
    const int b     = blockIdx.z;
    const int gx0   = xbase - RADIUS;
    const int gy0   = ybase - RADIUS;

    const float* gplanes = guidance + (size_t)b * GCH * H * W;
    const float* iplanes = image    + (size_t)b * ICH * H * W;

    // Interior tiles: halo (and the 32-wide padded TDM read) fully in-bounds.
    bool interior = (xbase >= TILE) && (xbase <= W - 2 * TILE) &&
                    (ybase >= TILE) && (ybase <= H - 2 * TILE);

#if HAVE_TDM
    if (interior) {
        const unsigned lds_base = (unsigned)(size_t)(void*)smem; // low 32b = LDS byte offset
        const int wave = __builtin_amdgcn_readfirstlane(tid >> 5);
        for (int plane = wave; plane < NPLANES; plane += 8) {
            const float* src = (plane < GCH)
                ? (gplanes + (size_t)plane * (H * W))
                : (iplanes + (size_t)(plane - GCH) * (H * W));
            const uint64_t ga = (uint64_t)(uintptr_t)(src + (size_t)gy0 * W + gx0);
            const unsigned lds_addr = lds_base + (unsigned)plane * (PLANE_ELEMS * 4u);

            // ---- D# group 0 (128b): count=1, lds_addr, global_addr, type=2
            v4u g0;
            g0.x = 1u;
            g0.y = lds_addr;
            g0.z = (unsigned)(ga & 0xFFFFFFFFu);
            g0.w = ((unsigned)(ga >> 32) & 0x01FFFFFFu) | (2u << 30);

            // ---- D# group 1 (256b): data_size=4B, tensor 256x256, tile 32x26,
            //      tensor_dim0_stride = 256 elements.
            v8i g1;
            g1[0] = 0x00020000;                        // wg_mask=0, data_size=2 (4B)
            g1[1] = (int)((unsigned)W << 16);          // tensor_dim0[15:0]=256
            g1[2] = (int)((unsigned)W << 16);          // tensor_dim1[15:0]=256
            g1[3] = (int)((unsigned)LDS_STRIDE << 16); // tile_dim0=32
            g1[4] = HALO;                              // tile_dim1=26, tile_dim2=0
            g1[5] = W;                                 // tensor_dim0_stride=256
            g1[6] = 0;
            g1[7] = 0;

            v4i gz4 = {0, 0, 0, 0};             // groups 2/3 unused (2D tile, count=1)
            v8i gz8 = {0, 0, 0, 0, 0, 0, 0, 0}; // extra operand (6-arg form), unused
            __builtin_amdgcn_tensor_load_to_lds(g0, g1, gz4, gz4, gz8, 0);
        }
        __builtin_amdgcn_s_wait_tensorcnt(0);
        __syncthreads(); // all waves' TDM data visible

        // Pre-scale the 8 guidance planes in place (each element reused ~121x).
        for (int idx = tid; idx < GCH * HALO * HALO; idx += 256) {
            const int plane = idx / (HALO * HALO);
            const int rem   = idx - plane * (HALO * HALO);
            const int r     = rem / HALO;
            const int c     = rem - r * HALO;
            const int off   = plane * PLANE_ELEMS + r * LDS_STRIDE + c;
            smem[off] *= kFscale[plane];
        }
    }
#else
    interior = false;
#endif

    if (!interior) {
        // Border tiles: manual gather with reflect padding; guidance scaled inline.
        for (int idx = tid; idx < NPLANES * HALO * HALO; idx += 256) {
            const int plane = idx / (HALO * HALO);
            const int rem   = idx - plane * (HALO * HALO);
            const int r     = rem / HALO;
            const int c     = rem - r * HALO;
            const int gy    = reflect_idx(gy0 + r);
            const int gx    = reflect_idx(gx0 + c);
            float v;
            if (plane < GCH) {
                v = gplanes[plane * (H * W) + gy * W + gx] * kFscale[plane];
            } else {
                v = iplanes[(plane - GCH) * (H * W) + gy * W + gx];
            }
            smem[plane * PLANE_ELEMS + r * LDS_STRIDE + c] = v;
        }
    }
    __syncthreads();

    // Center (pre-scaled) guidance vector for this pixel.
    float gc[GCH];
    const int ccenter = (ty + RADIUS) * LDS_STRIDE + (tx + RADIUS);
#pragma unroll
    for (int c = 0; c < GCH; ++c)
        gc[c] = smem[c * PLANE_ELEMS + ccenter];

    float a0 = 0.0f, a1 = 0.0f, a2 = 0.0f, sw = 0.0f;

#pragma unroll 1
    for (int dy = 0; dy < KSIZE; ++dy) {
        const int rowoff = (ty + dy) * LDS_STRIDE + tx;
#pragma unroll
        for (int dx = 0; dx < KSIZE; ++dx) {
            const int off = rowoff + dx;
            float maha = 0.0f;
#pragma unroll
            for (int c = 0; c < GCH; ++c) {
                const float d = smem[c * PLANE_ELEMS + off] - gc[c];
                maha = fmaf(d, d, maha);
            }
            // exp(-0.5*sum sinv_c*diff^2) == exp2(-maha); neg is a free src modifier
            const float wgt = exp2f(-maha);
            a0 = fmaf(wgt, smem[(GCH + 0) * PLANE_ELEMS + off], a0);
            a1 = fmaf(wgt, smem[(GCH + 1) * PLANE_ELEMS + off], a1);
            a2 = fmaf(wgt, smem[(GCH + 2) * PLANE_ELEMS + off], a2);
            sw += wgt;
        }
    }

    const float inv = 1.0f / fmaxf(sw, 1e-10f);
    const int y = ybase + ty;
    const int x = xbase + tx;
    float* ob = out + (size_t)b * ICH * H * W + y * W + x;
    ob[0 * H * W] = a0 * inv;
    ob[1 * H * W] = a1 * inv;
    ob[2 * H * W] = a2 * inv;
}

extern "C" void kernel_launch(void* const* d_in, const int* in_sizes, int n_in,
                              void* d_out, int out_size, void* d_ws, size_t ws_size,
                              hipStream_t stream) {
    const float* image    = (const float*)d_in[0]; // (B,3,256,256) fp32
    const float* guidance = (const float*)d_in[1]; // (B,8,256,256) fp32
    float* out = (float*)d_out;                    // (B,3,256,256) fp32

    const int B = in_sizes[0] / (ICH * H * W);
    dim3 grid(W / TILE, H / TILE, B);
    jbf_kernel<<<grid, dim3(256), 0, stream>>>(image, guidance, out);
}